// LiquidRON_84516366450724
// MI455X (gfx1250) — compile-verified
//
#include <hip/hip_runtime.h>

#define NTMAX 1024
#define DCHUNK 1024   // floats of data[] staged per async chunk (4 KB)

// ---------------------------------------------------------------------------
// One-time transpose: ST[j*n + i] = S[i*n + j]  (makes fired-column gathers
// coalesced in the persistent kernel: lane i reads consecutive addresses).
// ---------------------------------------------------------------------------
__global__ void izh_transpose_kernel(const float* __restrict__ S,
                                     float* __restrict__ ST, int n) {
  __shared__ float tile[32][33];
  int x = blockIdx.x * 32 + threadIdx.x;
  int y = blockIdx.y * 32 + threadIdx.y;
  if (x < n && y < n) tile[threadIdx.y][threadIdx.x] = S[(size_t)y * n + x];
  __syncthreads();
  int tx = blockIdx.y * 32 + threadIdx.x;
  int ty = blockIdx.x * 32 + threadIdx.y;
  if (tx < n && ty < n) ST[(size_t)ty * n + tx] = tile[threadIdx.x][threadIdx.y];
}

// ---------------------------------------------------------------------------
// Persistent single-workgroup Izhikevich scan.
// 1024 threads (32 wave32 waves) on one WGP; neuron i owned by thread i.
// Per step: ballot fired bits -> LDS masks -> wave0 builds a compacted,
// deterministic (ascending) list of PRE-SCALED element offsets (j*js) ->
// every thread accumulates I = x_t*U[i] + sum over fired columns (coalesced
// L2-resident loads, 4 accumulators for MLP), then the pointwise update.
// data[] is double-buffer staged into LDS with gfx1250 async global->LDS
// loads (ASYNCcnt / s_wait_asynccnt); columns are software-prefetched.
// ---------------------------------------------------------------------------
__global__ __launch_bounds__(1024, 1)
void izh_scan_kernel(const float* __restrict__ data,
                     const float* __restrict__ S,
                     const float* __restrict__ ST,
                     int use_st,
                     const float* __restrict__ a,
                     const float* __restrict__ b,
                     const float* __restrict__ c,
                     const float* __restrict__ d,
                     const float* __restrict__ U,
                     const float* __restrict__ v0,
                     const float* __restrict__ u0,
                     float* __restrict__ out,
                     int T, int N)
{
  __shared__ unsigned s_masks[NTMAX / 32];
  __shared__ alignas(16) unsigned s_idx[NTMAX];   // pre-scaled element offsets
  __shared__ int s_nnz;
  __shared__ float s_chunk[2][DCHUNK];

  const int i      = threadIdx.x;
  const int lane   = i & 31;
  const int wid    = i >> 5;
  const int nwaves = (int)(blockDim.x >> 5);
  const bool act     = (i < N);
  const bool staging = (nwaves >= 2);

  float v = 0.f, u = 0.f, an = 0.f, bn = 0.f, cn = 0.f, dn = 0.f, Un = 0.f;
  if (act) {
    v = v0[i]; u = u0[i];
    an = a[i]; bn = b[i]; cn = c[i]; dn = d[i]; Un = U[i];
  }

  float* o_st = out;                       // states  [T*N]
  float* o_v  = out + (size_t)T * N;       // v final [N]
  float* o_u  = o_v + N;                   // u final [N]
  float* o_fi = o_u + N;                   // fired_hist [T*N]

  // Per-thread base pointer; fired column j lives at colbase[j * js].
  const float* colbase = use_st ? (ST + i) : (S + (size_t)i * (size_t)N);
  const unsigned jsu   = use_st ? (unsigned)N : 1u;   // scale applied by wave0

  // Prologue: wave1 asynchronously stages data chunk 0 into LDS.
  if (staging && wid == 1 && DCHUNK <= T) {
    const float* src = data;
    unsigned lb = (unsigned)(size_t)&s_chunk[0][0];
#pragma unroll
    for (int seg = 0; seg < (DCHUNK * 4) / 512; ++seg) {
      unsigned vo = (unsigned)(lane * 16 + seg * 512);
      unsigned lv = lb + vo;
      asm volatile("global_load_async_to_lds_b128 %0, %1, %2 offset:0"
                   :: "v"(lv), "v"(vo), "s"(src) : "memory");
    }
  }

  for (int t = 0; t < T; ++t) {
    // ---- publish this step's fired bits (pre-update spikes) --------------
    bool fired = act && (v >= 30.0f);
    unsigned long long bal = __ballot(fired);
    if (lane == 0) s_masks[wid] = (unsigned)bal;
    __syncthreads();  // (1) masks visible

    // ---- wave0: deterministic compaction into pre-scaled offsets ---------
    if (wid == 0) {
      unsigned m = (lane < nwaves) ? s_masks[lane] : 0u;
      int cnt  = __popc(m);
      int incl = cnt;
#pragma unroll
      for (int o = 1; o < 32; o <<= 1) {
        int up = __shfl_up(incl, o, 32);
        if (lane >= o) incl += up;
      }
      int off = incl - cnt;                 // exclusive prefix
      if (lane == 31) s_nnz = incl;         // total fired count
      unsigned base = (unsigned)(lane << 5);
      while (m) {
        int bpos = __builtin_ctz(m);
        m &= m - 1;
        s_idx[off++] = (base + (unsigned)bpos) * jsu;  // scale once, here
      }
    }

    // ---- wave1: async data staging bookkeeping (once per 1024 steps) -----
    if (staging && wid == 1 && (t & (DCHUNK - 1)) == 0) {
      asm volatile("s_wait_asynccnt 0" ::: "memory");   // current chunk ready
      int next = (t / DCHUNK) + 1;
      if ((size_t)(next + 1) * DCHUNK <= (size_t)T) {   // only full chunks
        const float* src = data + (size_t)next * DCHUNK;
        unsigned lb = (unsigned)(size_t)&s_chunk[next & 1][0];
#pragma unroll
        for (int seg = 0; seg < (DCHUNK * 4) / 512; ++seg) {
          unsigned vo = (unsigned)(lane * 16 + seg * 512);
          unsigned lv = lb + vo;
          asm volatile("global_load_async_to_lds_b128 %0, %1, %2 offset:0"
                       :: "v"(lv), "v"(vo), "s"(src) : "memory");
        }
      }
    }
    __syncthreads();  // (2) offset list + staged data visible

    // ---- per-neuron work --------------------------------------------------
    if (act) {
      const int nn = s_nnz;
      bool have = staging && (((t & ~(DCHUNK - 1)) + DCHUNK) <= T);
      float xt = have ? s_chunk[(t / DCHUNK) & 1][t & (DCHUNK - 1)] : data[t];

      if (fired) { v = cn; u += dn; }       // reset + recovery bump

      // I = x_t*U + sum over fired columns; 4 independent accumulators.
      float a0 = xt * Un, a1 = 0.f, a2 = 0.f, a3 = 0.f;
      const uint4* idx4 = reinterpret_cast<const uint4*>(s_idx);
      int k = 0;
      for (; k + 4 <= nn; k += 4) {
        uint4 q = idx4[k >> 2];             // one ds_load_b128: 4 offsets
        if (k + 8 < nn)
          __builtin_prefetch(&colbase[s_idx[k + 8]], 0, 0);  // global_prefetch
        a0 += colbase[q.x];
        a1 += colbase[q.y];
        a2 += colbase[q.z];
        a3 += colbase[q.w];
      }
      for (; k < nn; ++k) a0 += colbase[s_idx[k]];
      float I = (a0 + a1) + (a2 + a3);

      // Izhikevich update (single half-step, matching the reference)
      v = v + 0.5f * (0.04f * v * v + 5.0f * v + 140.0f - u + I);
      u = u + an * (bn * v - u);

      size_t o = (size_t)t * (size_t)N + (size_t)i;
      o_st[o] = (v >= 30.0f) ? 1.0f : 0.0f;   // states[t]
      o_fi[o] = fired ? 1.0f : 0.0f;          // fired_hist[t]
    }
  }

  if (act) { o_v[i] = v; o_u[i] = u; }
}

// ---------------------------------------------------------------------------
extern "C" void kernel_launch(void* const* d_in, const int* in_sizes, int n_in,
                              void* d_out, int out_size, void* d_ws, size_t ws_size,
                              hipStream_t stream) {
  const float* data = (const float*)d_in[0];
  const float* S    = (const float*)d_in[1];
  const float* a    = (const float*)d_in[2];
  const float* b    = (const float*)d_in[3];
  const float* c    = (const float*)d_in[4];
  const float* d    = (const float*)d_in[5];
  const float* U    = (const float*)d_in[6];
  const float* v0   = (const float*)d_in[7];
  const float* u0   = (const float*)d_in[8];
  const int T = in_sizes[0];
  const int N = in_sizes[2];
  float* out = (float*)d_out;

  float* ST = (float*)d_ws;
  const int use_st = (ws_size >= (size_t)N * (size_t)N * sizeof(float)) ? 1 : 0;
  if (use_st) {
    dim3 grid((N + 31) / 32, (N + 31) / 32), blk(32, 32);
    hipLaunchKernelGGL(izh_transpose_kernel, grid, blk, 0, stream, S, ST, N);
  }

  int nthreads = ((N + 31) / 32) * 32;
  if (nthreads < 64) nthreads = 64;
  if (nthreads > 1024) nthreads = 1024;
  hipLaunchKernelGGL(izh_scan_kernel, dim3(1), dim3(nthreads), 0, stream,
                     data, S, ST, use_st, a, b, c, d, U, v0, u0, out, T, N);
}